// FeaturePropogation_47287589929775
// MI455X (gfx1250) — compile-verified
//
#include <hip/hip_runtime.h>
#include <hip/hip_bf16.h>

#define EPSF 1e-5f
#define C1 64
#define C2 128
#define NS 16
#define WAVES_PER_WG 4
#define PTS_PER_WAVE 8
#define PTS_PER_WG (WAVES_PER_WG * PTS_PER_WAVE)

typedef float v2f __attribute__((ext_vector_type(2)));
typedef float v8f __attribute__((ext_vector_type(8)));

// ---------------------------------------------------------------------------
// Pass 1: async-gather 16 neighbor rows (16x128 f32 tile) into LDS with
// double buffering (ASYNCcnt), fp32 WMMA GEMM vs W^T staged in LDS, bias add,
// per-(point,channel) max/min over neighbors, per-channel sum/sumsq for BN1.
// ---------------------------------------------------------------------------
__global__ __launch_bounds__(128)
void gather_gemm_pool_kernel(const float* __restrict__ f2,
                             const int*   __restrict__ idx,
                             const float* __restrict__ W,     // [C1][C2]
                             const float* __restrict__ bias,  // [C1]
                             float* __restrict__ hmax,        // [m][C1]
                             float* __restrict__ hmin,        // [m][C1]
                             float* __restrict__ stats1,      // [0..63]=sum [64..127]=sumsq
                             int m)
{
    __shared__ float sW[C1 * C2];                         // 32 KB
    __shared__ float sTile[WAVES_PER_WG][2][NS * C2];     // 4 * 2 * 8 KB = 64 KB
    __shared__ float sSum[C1];
    __shared__ float sSumSq[C1];

    const int tid = threadIdx.x;

    // Stage W into LDS (float4 vectorized)
    {
        const float4* Wv  = (const float4*)W;
        float4*       sWv = (float4*)sW;
        #pragma unroll
        for (int i = tid; i < (C1 * C2) / 4; i += 128) sWv[i] = Wv[i];
    }
    if (tid < C1) { sSum[tid] = 0.f; sSumSq[tid] = 0.f; }
    __syncthreads();

    const int wave  = tid >> 5;
    const int lane  = tid & 31;          // wave32
    const int lrow  = lane & 15;         // M row (A) / N col (B) within half-wave
    const int khalf = (lane >> 4) << 1;  // K sub-offset: 0 (lanes 0-15), 2 (lanes 16-31)

    // Bias fragment per N-tile (channel = t*16 + lrow)
    float bfrag[4];
    #pragma unroll
    for (int t = 0; t < 4; ++t) bfrag[t] = bias[t * 16 + lrow];

    const int pbase = blockIdx.x * PTS_PER_WG + wave * PTS_PER_WAVE;
    int pcnt = m - pbase;
    if (pcnt > PTS_PER_WAVE) pcnt = PTS_PER_WAVE;

    const unsigned ldsTile0 = (unsigned)(uintptr_t)(&sTile[wave][0][0]);
    const unsigned ldsTile1 = (unsigned)(uintptr_t)(&sTile[wave][1][0]);

    // Stage one point's gathered 16x128 tile into an LDS buffer: one async
    // b128 per row (32 lanes x 16B = full 512B row), tracked by ASYNCcnt.
    auto stage = [&](int p, unsigned ldsbase) {
        #pragma unroll
        for (int r = 0; r < NS; ++r) {
            const int row = idx[(size_t)p * NS + r];            // uniform per wave
            const float* gsrc = f2 + (size_t)row * C2 + lane * 4;
            const unsigned ldst = ldsbase + (unsigned)(r * (C2 * 4) + lane * 16);
            asm volatile("global_load_async_to_lds_b128 %0, %1, off"
                         :: "v"(ldst), "v"(gsrc)
                         : "memory");
        }
    };

    if (pcnt > 0) stage(pbase, ldsTile0);

    for (int pi = 0; pi < pcnt; ++pi) {
        const int cur = pi & 1;
        if (pi + 1 < pcnt) {
            stage(pbase + pi + 1, cur ? ldsTile0 : ldsTile1);
            // 32 async ops in flight; in-order completion => cnt<=16 means the
            // current buffer's 16 row-copies are done, next 16 still in flight.
            asm volatile("s_wait_asynccnt 0x10" ::: "memory");
        } else {
            asm volatile("s_wait_asynccnt 0x0" ::: "memory");
        }

        const int p = pbase + pi;
        const float* atile = &sTile[wave][cur][0];
        const float* abase = atile + lrow * C2 + khalf;

        v8f acc0 = {}, acc1 = {}, acc2 = {}, acc3 = {};

        #pragma unroll 4
        for (int k = 0; k < C2; k += 4) {
            // A fragment from LDS tile: lane holds row[k + khalf .. +1]
            v2f a = *(const v2f*)(abase + k);
            // B fragments: W[n][k + khalf .. +1], n = t*16 + lrow (ds_load_b64)
            const float* wb = &sW[lrow * C2 + k + khalf];
            v2f b0 = *(const v2f*)(wb + 0 * 16 * C2);
            v2f b1 = *(const v2f*)(wb + 1 * 16 * C2);
            v2f b2 = *(const v2f*)(wb + 2 * 16 * C2);
            v2f b3 = *(const v2f*)(wb + 3 * 16 * C2);
            acc0 = __builtin_amdgcn_wmma_f32_16x16x4_f32(false, a, false, b0, (short)0, acc0, false, false);
            acc1 = __builtin_amdgcn_wmma_f32_16x16x4_f32(false, a, false, b1, (short)0, acc1, false, false);
            acc2 = __builtin_amdgcn_wmma_f32_16x16x4_f32(false, a, false, b2, (short)0, acc2, false, false);
            acc3 = __builtin_amdgcn_wmma_f32_16x16x4_f32(false, a, false, b3, (short)0, acc3, false, false);
        }

        // Post-process: bias, max/min over neighbors, channel stats.
        v8f accs[4] = { acc0, acc1, acc2, acc3 };
        #pragma unroll
        for (int t = 0; t < 4; ++t) {
            const float bv = bfrag[t];
            float mx = -3.402823466e38f, mn = 3.402823466e38f;
            float s = 0.f, ss = 0.f;
            #pragma unroll
            for (int j = 0; j < 8; ++j) {
                float h = accs[t][j] + bv;   // neighbor M = j (+8 in upper half)
                mx = fmaxf(mx, h);
                mn = fminf(mn, h);
                s  += h;
                ss += h * h;
            }
            // Combine the two half-waves (same channel, neighbors 0-7 vs 8-15)
            mx = fmaxf(mx, __shfl_xor(mx, 16, 32));
            mn = fminf(mn, __shfl_xor(mn, 16, 32));
            s  += __shfl_xor(s, 16, 32);
            ss += __shfl_xor(ss, 16, 32);
            if (lane < 16) {
                const size_t o = (size_t)p * C1 + t * 16 + lrow;
                hmax[o] = mx;
                hmin[o] = mn;
                atomicAdd(&sSum[t * 16 + lrow], s);
                atomicAdd(&sSumSq[t * 16 + lrow], ss);
            }
        }
    }

    __syncthreads();
    if (tid < C1)            atomicAdd(&stats1[tid], sSum[tid]);
    else if (tid < 2 * C1)   atomicAdd(&stats1[tid], sSumSq[tid - C1]);
}

// ---------------------------------------------------------------------------
// Pass 2: finalize BN1 per-channel affine, monotone max/min selection, ReLU,
// residual add of f1; write out and accumulate BN2 statistics.
// ---------------------------------------------------------------------------
__global__ __launch_bounds__(256)
void bn1_pool_residual_kernel(const float* __restrict__ f1,
                              const float* __restrict__ hmax,
                              const float* __restrict__ hmin,
                              const float* __restrict__ stats1,
                              const float* __restrict__ gamma1,
                              const float* __restrict__ beta1,
                              float* __restrict__ out,
                              float* __restrict__ stats2,
                              int m)
{
    __shared__ float s2[C1], ss2[C1];
    const int tid = threadIdx.x;
    if (tid < C1) { s2[tid] = 0.f; ss2[tid] = 0.f; }
    __syncthreads();

    const long i = (long)blockIdx.x * 256 + tid;
    const long total = (long)m * C1;
    if (i < total) {
        const int d = (int)(i & (C1 - 1));
        const float cnt  = (float)m * (float)NS;
        const float mean = stats1[d] / cnt;
        const float var  = stats1[C1 + d] / cnt - mean * mean;
        const float a    = gamma1[d] * rsqrtf(var + EPSF);
        const float sh   = beta1[d] - a * mean;
        // BN1+ReLU is monotone in h for fixed sign(a): pooled value from max or min
        const float v = (a >= 0.f) ? hmax[i] : hmin[i];
        const float r = fmaxf(a * v + sh, 0.f);
        const float o = f1[i] + r;
        out[i] = o;
        atomicAdd(&s2[d], o);
        atomicAdd(&ss2[d], o * o);
    }
    __syncthreads();
    if (tid < C1)          atomicAdd(&stats2[tid], s2[tid]);
    else if (tid < 2 * C1) atomicAdd(&stats2[tid], ss2[tid - C1]);
}

// ---------------------------------------------------------------------------
// Pass 3: BN2 normalize in place.
// ---------------------------------------------------------------------------
__global__ __launch_bounds__(256)
void bn2_kernel(const float* __restrict__ stats2,
                const float* __restrict__ gamma2,
                const float* __restrict__ beta2,
                float* __restrict__ out,
                int m)
{
    const long i = (long)blockIdx.x * 256 + threadIdx.x;
    const long total = (long)m * C1;
    if (i >= total) return;
    const int d = (int)(i & (C1 - 1));
    const float mean = stats2[d] / (float)m;
    const float var  = stats2[C1 + d] / (float)m - mean * mean;
    out[i] = gamma2[d] * (out[i] - mean) * rsqrtf(var + EPSF) + beta2[d];
}

// ---------------------------------------------------------------------------
extern "C" void kernel_launch(void* const* d_in, const int* in_sizes, int n_in,
                              void* d_out, int out_size, void* d_ws, size_t ws_size,
                              hipStream_t stream)
{
    const float* f1     = (const float*)d_in[0];
    const float* f2     = (const float*)d_in[1];
    const int*   idx    = (const int*)  d_in[2];
    const float* W      = (const float*)d_in[3];
    const float* bias   = (const float*)d_in[4];
    const float* gamma1 = (const float*)d_in[5];
    const float* beta1  = (const float*)d_in[6];
    const float* gamma2 = (const float*)d_in[7];
    const float* beta2  = (const float*)d_in[8];

    const int m = in_sizes[0] / C1;   // 100000

    // Workspace layout
    float* stats1 = (float*)d_ws;                       // 128 floats
    float* stats2 = stats1 + 128;                       // 128 floats
    float* hmax   = (float*)((char*)d_ws + 1024);       // m*C1 floats
    float* hmin   = hmax + (size_t)m * C1;              // m*C1 floats

    // Zero the stats accumulators (graph-capture-safe)
    hipMemsetAsync(d_ws, 0, 1024, stream);

    const int g1 = (m + PTS_PER_WG - 1) / PTS_PER_WG;
    gather_gemm_pool_kernel<<<g1, dim3(128), 0, stream>>>(f2, idx, W, bias, hmax, hmin, stats1, m);

    const long total = (long)m * C1;
    const int g2 = (int)((total + 255) / 256);
    bn1_pool_residual_kernel<<<g2, dim3(256), 0, stream>>>(f1, hmax, hmin, stats1,
                                                           gamma1, beta1, (float*)d_out, stats2, m);
    bn2_kernel<<<g2, dim3(256), 0, stream>>>(stats2, gamma2, beta2, (float*)d_out, m);
}